// Gaussian_Forward_Model_27616639714071
// MI455X (gfx1250) — compile-verified
//
#include <hip/hip_runtime.h>

// MI455X / gfx1250: wave32, WMMA (not MFMA).
// 3x3 Gaussian conv, padding=1, over [64,1,512,512] fp32.
// Memory-bound (134 MB @ 23.3 TB/s ~ 6us floor) -> keep fp32, feed the
// matrix pipe with V_WMMA_F32_16X16X4_F32 so the 9-tap MACs come off the VALU.
//
// Formulation per 16x16 output tile:
//   C += sum_{s=0..2} W_s (16x16, tridiagonal from kernel column s)
//                   * X_s (16x16, input rows r0..r0+15, cols shifted by s-1)
// K=16 split into four K=4 WMMA chunks -> 12 wmma per tile.
// Band terms touching tile rows -1 and 16 are scalar corrections on
// accumulator rows 0 and 15.

typedef __attribute__((ext_vector_type(2))) float v2f;
typedef __attribute__((ext_vector_type(8))) float v8f;

#define HW 512

// Zero-padded load: clamp the index so the load is always legal, then
// select 0 for out-of-range. No EXEC divergence -> EXEC stays all-ones
// for the WMMAs (ISA requirement).
__device__ __forceinline__ float ldz(const float* __restrict__ img, int r, int c) {
    int rc = min(max(r, 0), HW - 1);
    int cc = min(max(c, 0), HW - 1);
    float v = img[rc * HW + cc];
    return (r == rc && c == cc) ? v : 0.0f;
}

// Tridiagonal band matrix entry: W_s[m][j] = k[j-m+1][s]
__device__ __forceinline__ float bandw(int m, int j, float k0, float k1, float k2) {
    int d = j - m;
    return d == -1 ? k0 : d == 0 ? k1 : d == 1 ? k2 : 0.0f;
}

__global__ __launch_bounds__(256)
void gauss3x3_wmma_kernel(const float* __restrict__ in,
                          const float* __restrict__ weight,
                          const float* __restrict__ wfac,
                          float* __restrict__ out)
{
    const int lane  = threadIdx.x & 31;
    const int wid   = threadIdx.x >> 5;        // wave id within block (8 waves)
    const int tile  = blockIdx.x * 8 + wid;    // 65536 tiles total
    const int img   = tile >> 10;              // 1024 tiles per 512x512 image
    const int trem  = tile & 1023;
    const int r0    = (trem >> 5) << 4;        // tile row origin
    const int c0    = (trem & 31) << 4;        // tile col origin

    const int  m     = lane & 15;              // M (A/C row) and N (col) index
    const bool hi    = lane >= 16;
    const int  krow0 = hi ? 2 : 0;             // K offset of this half-wave in each chunk
    const int  gc    = c0 + m;                 // global column for B/C/D

    const float* ip = in + (size_t)img * (HW * HW);

    // ---- Reproduce reference STE weight quantization (forward value) ----
    // wc = clip(w, 0.001, 0.999); wf = clip(wf, 1.001, 254.999);
    // wr = max(round_half_even(wc*wf), 0.001)
    float k[3][3];
    {
        float wf = fminf(fmaxf(wfac[0], 1.001f), 254.999f);
#pragma unroll
        for (int i = 0; i < 9; ++i) {
            float w = fminf(fmaxf(weight[i], 0.001f), 0.999f);
            float q = rintf(w * wf);           // jnp.round = round-half-to-even
            k[i / 3][i % 3] = fmaxf(q, 0.001f);
        }
    }

    // ---- Gather the 27 input samples this lane contributes to B + corrections.
    // B layout (4x16 f32, mirrors documented 16x4 A layout):
    //   chunk c VGPR0: lanes0-15 -> K=4c+0, lanes16-31 -> K=4c+2
    //   chunk c VGPR1: lanes0-15 -> K=4c+1, lanes16-31 -> K=4c+3
    float xv[3][8];
    float xc[3];
    const int rcorr = r0 + (hi ? 16 : -1);     // out-of-tile band row
#pragma unroll
    for (int s = 0; s < 3; ++s) {
        const int cc = gc + s - 1;
#pragma unroll
        for (int t = 0; t < 8; ++t) {
            const int row = r0 + (t >> 1) * 4 + krow0 + (t & 1);
            xv[s][t] = ldz(ip, row, cc);
        }
        xc[s] = ldz(ip, rcorr, cc);
    }

    // ---- 12 x V_WMMA_F32_16X16X4_F32 ----
    v8f acc = {};
#pragma unroll
    for (int s = 0; s < 3; ++s) {
        const float kc0 = k[0][s], kc1 = k[1][s], kc2 = k[2][s];
#pragma unroll
        for (int c = 0; c < 4; ++c) {
            const int j0 = 4 * c + krow0;
            v2f a, b;
            a[0] = bandw(m, j0,     kc0, kc1, kc2);   // A VGPR0
            a[1] = bandw(m, j0 + 1, kc0, kc1, kc2);   // A VGPR1
            b[0] = xv[s][2 * c + 0];                  // B VGPR0
            b[1] = xv[s][2 * c + 1];                  // B VGPR1
            acc = __builtin_amdgcn_wmma_f32_16x16x4_f32(
                /*neg_a=*/false, a, /*neg_b=*/false, b,
                /*c_mod=*/(short)0, acc,
                /*reuse_a=*/false, /*reuse_b=*/false);
        }
    }

    // ---- Boundary band corrections: row 0 needs k[0][*]*x[r0-1], row 15 needs
    // k[2][*]*x[r0+16]. C/D layout: VGPR v = row v (lanes 0-15) / v+8 (16-31).
    {
        const float kr0 = hi ? k[2][0] : k[0][0];
        const float kr1 = hi ? k[2][1] : k[0][1];
        const float kr2 = hi ? k[2][2] : k[0][2];
        const float corr = kr0 * xc[0] + kr1 * xc[1] + kr2 * xc[2];
        acc[0] += hi ? 0.0f : corr;   // tile row 0  (lanes 0-15)
        acc[7] += hi ? corr : 0.0f;   // tile row 15 (lanes 16-31)
    }

    // ---- Store: output shape [1,64,1,512,512] == input layout ----
    float* op = out + (size_t)img * (HW * HW);
    const int rowbase = r0 + (hi ? 8 : 0);
#pragma unroll
    for (int v = 0; v < 8; ++v)
        op[(size_t)(rowbase + v) * HW + gc] = acc[v];
}

extern "C" void kernel_launch(void* const* d_in, const int* in_sizes, int n_in,
                              void* d_out, int out_size, void* d_ws, size_t ws_size,
                              hipStream_t stream) {
    const float* in  = (const float*)d_in[0];   // [64,1,512,512] f32
    const float* w   = (const float*)d_in[1];   // [1,9] f32
    const float* wf  = (const float*)d_in[2];   // [1,1] f32
    float*       out = (float*)d_out;           // [1,64,1,512,512] f32

    const int tiles  = 64 * 32 * 32;            // 65536 16x16 tiles
    dim3 grid(tiles / 8);                       // 8 waves (tiles) per block
    dim3 block(256);
    gauss3x3_wmma_kernel<<<grid, block, 0, stream>>>(in, w, wf, out);
}